// InfoNCELossBG_16758962389002
// MI455X (gfx1250) — compile-verified
//
#include <hip/hip_runtime.h>
#include <math.h>

typedef __attribute__((ext_vector_type(2))) float v2f;
typedef __attribute__((ext_vector_type(8))) float v8f;

#define B_DIM 256
#define K_DIM 512
#define D_DIM 1024
#define INV_T 14.285714285714286f /* 1 / 0.07 */

// Block-wide sum for 256 threads (8 wave32). `red` must hold 8 floats.
__device__ __forceinline__ float block_reduce_sum(float v, float* red) {
#pragma unroll
    for (int off = 16; off >= 1; off >>= 1)
        v += __shfl_xor(v, off, 32);
    const int wave = threadIdx.x >> 5;
    if ((threadIdx.x & 31) == 0) red[wave] = v;
    __syncthreads();
    float total = 0.0f;
#pragma unroll
    for (int i = 0; i < 8; ++i) total += red[i];
    __syncthreads();
    return total;
}

// One block per batch b. 256 threads = 8 waves; each wave owns 4 K-tiles of 16 rows.
__global__ __launch_bounds__(256)
void infonce_bg_main(const float* __restrict__ bg_img,
                     const float* __restrict__ fg_pro,
                     const float* __restrict__ bg_pro,
                     float* __restrict__ acc /* acc[0]=sum exp(bg), acc[1]=sum exp(fg) */) {
    __shared__ float xs[D_DIM];   // normalized image feature for this batch
    __shared__ float red[8];

    const int b   = blockIdx.x;
    const int tid = threadIdx.x;
    const float* img = bg_img + (size_t)b * D_DIM;

    // ---- Phase 1: normalize bg_img[b] into LDS ----
    float p = 0.0f;
#pragma unroll
    for (int d = tid; d < D_DIM; d += 256) {
        float v = img[d];
        p += v * v;
    }
    const float n2 = block_reduce_sum(p, red);
    const float inv_norm = 1.0f / sqrtf(n2);
#pragma unroll
    for (int d = tid; d < D_DIM; d += 256)
        xs[d] = img[d] * inv_norm;
    __syncthreads();

    // ---- Phase 2: foreground logit -> exp -> global accumulator ----
    const float* fg = fg_pro + (size_t)b * D_DIM;
    float q = 0.0f;
#pragma unroll
    for (int d = tid; d < D_DIM; d += 256)
        q += xs[d] * fg[d];
    const float fgdot = block_reduce_sum(q, red);
    if (tid == 0)
        atomicAdd(acc + 1, __expf(fgdot * INV_T));

    // ---- Phase 3: background logits via V_WMMA_F32_16X16X4_F32 ----
    // A tile = 16 K-rows x 4 D-cols of bg_pro[b]; B = x_b broadcast to all 16 columns.
    // f32 16x4 A-fragment layout: V0 = {K0 | K2}, V1 = {K1 | K3} over lane halves,
    // so lane (m, hi) holds A[m][d + 2*hi] and A[m][d + 1 + 2*hi] -> one b64 load.
    const int wave = tid >> 5;
    const int lane = tid & 31;
    const int m    = lane & 15;
    const int hi   = lane >> 4;

    float wavesum = 0.0f;
#pragma unroll
    for (int t = 0; t < 4; ++t) {
        const int k0 = (wave + 8 * t) * 16;
        const float* rowA = bg_pro + (((size_t)b * K_DIM) + k0 + m) * D_DIM + 2 * hi;
        const float* rowB = xs + 2 * hi;

        v8f acc0 = {};
        v8f acc1 = {};
#pragma unroll 4
        for (int d = 0; d < D_DIM; d += 8) {
            v2f a0 = *(const v2f*)(rowA + d);
            v2f a1 = *(const v2f*)(rowA + d + 4);
            v2f b0 = *(const v2f*)(rowB + d);
            v2f b1 = *(const v2f*)(rowB + d + 4);
            acc0 = __builtin_amdgcn_wmma_f32_16x16x4_f32(
                false, a0, false, b0, (short)0, acc0, false, false);
            acc1 = __builtin_amdgcn_wmma_f32_16x16x4_f32(
                false, a1, false, b1, (short)0, acc1, false, false);
        }

        // Every column of C is identical (B was broadcast), so summing
        // exp() over all 8 regs x 32 lanes gives 16x the tile's true sum.
        float s = 0.0f;
#pragma unroll
        for (int r = 0; r < 8; ++r)
            s += __expf((acc0[r] + acc1[r]) * INV_T);
        wavesum += s;
    }

#pragma unroll
    for (int off = 16; off >= 1; off >>= 1)
        wavesum += __shfl_xor(wavesum, off, 32);
    if (lane == 0)
        atomicAdd(acc + 0, wavesum * (1.0f / 16.0f));
}

__global__ void infonce_bg_final(const float* __restrict__ acc,
                                 float* __restrict__ out) {
    const float pos = acc[0] * (1.0f / (float)K_DIM); // mean over K
    const float neg = pos + acc[1];
    out[0] = -logf(pos / neg);
}

extern "C" void kernel_launch(void* const* d_in, const int* in_sizes, int n_in,
                              void* d_out, int out_size, void* d_ws, size_t ws_size,
                              hipStream_t stream) {
    const float* bg_img = (const float*)d_in[0]; // [B, D]
    const float* fg_pro = (const float*)d_in[1]; // [B, D]
    const float* bg_pro = (const float*)d_in[2]; // [B, K, D]
    float* out = (float*)d_out;
    float* acc = (float*)d_ws;

    hipMemsetAsync(acc, 0, 2 * sizeof(float), stream);
    infonce_bg_main<<<B_DIM, 256, 0, stream>>>(bg_img, fg_pro, bg_pro, acc);
    infonce_bg_final<<<1, 1, 0, stream>>>(acc, out);
}